// BatchNormSPD_89756226552163
// MI455X (gfx1250) — compile-verified
//
#include <hip/hip_runtime.h>
#include <math.h>

typedef __attribute__((ext_vector_type(2))) float v2f;
typedef __attribute__((ext_vector_type(8))) float v8f;

#define NMAT 16384   // N*H = 256*64 matrices of 32x32
#define NDIM 32

// workspace layout (float offsets).  All 32x32 blobs in ws except GACC use the same
// paired-row layout as LDS (only X / out / weight at the global ABI boundary
// and the GACC mean accumulator are row-major).
#define WS_GACC 0
#define WS_LACC 1024
#define WS_GSQ  2048
#define WS_GISQ 3072
#define WS_P    4096

// ---------------------------------------------------------------------------
// Paired-row LDS layout: element (r,c) at (r/2)*64 + 2c + (r&1).
//   B fragment (rows k,k+1, col n)  -> contiguous 8B pair  -> ds_load_b64
//     (two fragments merge into one ds_load_2addr_b64)
//   A fragment (row m, cols k,k+1)  -> stride-2 pair       -> ds_load_2addr_b32
//   C rows (2m,2m+1) at a column    -> contiguous 8B pair  -> ds_store_b64
// ---------------------------------------------------------------------------
__device__ __forceinline__ int mi(int r, int c) {
  return ((r >> 1) << 6) + (c << 1) + (r & 1);
}

enum EpiMode { EPI_STORE, EPI_GSTORE, EPI_ATOMIC };

// ---------------------------------------------------------------------------
// One wave: C = alpha*(A*B) (+ beta*I if DIAG), optionally B^T.
// A,B 32x32 paired-row LDS.  EPI_STORE: LDS paired-row; EPI_GSTORE: global
// row-major; EPI_ATOMIC: LDS atomic add.  The diagonal add is a separate
// per-lane fixup (lane i owns element (i,i)) so no select results live
// across the WMMA chains -> no spills.
// ---------------------------------------------------------------------------
template <bool TB, EpiMode EM, bool DIAG>
__device__ __forceinline__ void wave_mm32e(const float* A, const float* B,
                                           float* C, int lane,
                                           float alpha, float beta) {
  const int half = lane >> 4;
  const int l = lane & 15;

  v2f bf[8][2];
#pragma unroll
  for (int k0 = 0; k0 < 8; ++k0) {
    const int ka = k0 * 4 + 2 * half;   // even
#pragma unroll
    for (int tj = 0; tj < 2; ++tj) {
      const int n = tj * 16 + l;
      if (!TB) {
        bf[k0][tj] = *(const v2f*)(B + mi(ka, n));   // ds_load_b64
      } else {
        v2f t;
        t[0] = B[mi(n, ka)];
        t[1] = B[mi(n, ka + 1)];
        bf[k0][tj] = t;
      }
    }
  }

#pragma unroll
  for (int ti = 0; ti < 2; ++ti) {
    const int m = ti * 16 + l;
    v2f af[8];
#pragma unroll
    for (int k0 = 0; k0 < 8; ++k0) {
      const int ka = k0 * 4 + 2 * half;
      v2f t;
      t[0] = A[mi(m, ka)];
      t[1] = A[mi(m, ka + 1)];
      af[k0] = t;
    }
    v8f acc0 = {};
    v8f acc1 = {};
#pragma unroll
    for (int k0 = 0; k0 < 8; ++k0) {
      acc0 = __builtin_amdgcn_wmma_f32_16x16x4_f32(
          false, af[k0], false, bf[k0][0], (short)0, acc0, false, false);
      acc1 = __builtin_amdgcn_wmma_f32_16x16x4_f32(
          false, af[k0], false, bf[k0][1], (short)0, acc1, false, false);
    }
#pragma unroll
    for (int r = 0; r < 8; r += 2) {
      const int row = ti * 16 + r + 8 * half;     // even
      if (EM == EPI_STORE) {
        v2f p0, p1;
        p0[0] = alpha * acc0[r];  p0[1] = alpha * acc0[r + 1];
        p1[0] = alpha * acc1[r];  p1[1] = alpha * acc1[r + 1];
        *(v2f*)(C + mi(row, l))      = p0;        // ds_store_b64
        *(v2f*)(C + mi(row, 16 + l)) = p1;
      } else if (EM == EPI_GSTORE) {
        C[row * 32 + l]             = alpha * acc0[r];
        C[(row + 1) * 32 + l]       = alpha * acc0[r + 1];
        C[row * 32 + 16 + l]        = alpha * acc1[r];
        C[(row + 1) * 32 + 16 + l]  = alpha * acc1[r + 1];
      } else {
        atomicAdd(&C[mi(row, l)],          alpha * acc0[r]);
        atomicAdd(&C[mi(row, l) + 1],      alpha * acc0[r + 1]);
        atomicAdd(&C[mi(row, 16 + l)],     alpha * acc1[r]);
        atomicAdd(&C[mi(row, 16 + l) + 1], alpha * acc1[r + 1]);
      }
    }
  }
  if (DIAG) {
    // lane i fixes diagonal element (i,i); wave-private C, DS in-order
    C[mi(lane, lane)] += beta;
  }
}

template <bool TB>
__device__ __forceinline__ void wave_mm32(const float* A, const float* B,
                                          float* C, int lane) {
  wave_mm32e<TB, EPI_STORE, false>(A, B, C, lane, 1.0f, 0.0f);
}

// ---------------------------------------------------------------------------
// Coupled Newton-Schulz: A SPD -> Y ~ (A/c)^{1/2}, Z ~ (A/c)^{-1/2}, c=trace.
// T = (3I - ZY)/2 fused: alpha=-0.5, diag fixup +1.5.
// ---------------------------------------------------------------------------
__device__ __forceinline__ void wave_ns(const float* A, float*& Y, float*& Z,
                                        float*& T, float*& U, int lane,
                                        int iters, float& c_out) {
  float c = A[mi(lane, lane)];                 // butterfly trace reduction
#pragma unroll
  for (int m = 16; m >= 1; m >>= 1) c += __shfl_xor(c, m, 32);
  const float ic = 1.0f / c;
  for (int r = 0; r < 32; ++r) {
    Y[mi(r, lane)] = A[mi(r, lane)] * ic;
    Z[mi(r, lane)] = (r == lane) ? 1.0f : 0.0f;
  }
  for (int it = 0; it < iters; ++it) {
    wave_mm32e<false, EPI_STORE, true>(Z, Y, T, lane, -0.5f, 1.5f);
    wave_mm32<false>(Y, T, U, lane);
    { float* t = Y; Y = U; U = t; }
    wave_mm32<false>(T, Z, U, lane);
    { float* t = Z; Z = U; U = t; }
  }
  c_out = c;
}

// ---------------------------------------------------------------------------
__global__ void kZero(float* ws) {
  int i = blockIdx.x * blockDim.x + threadIdx.x;
  if (i < 2048) ws[i] = 0.0f;
}

// Arithmetic mean accumulator: 64 blocks x 256 matrices each (row-major acc).
__global__ void __launch_bounds__(256) kMean(const float* __restrict__ X,
                                             float* __restrict__ Gacc) {
  const int tid = threadIdx.x;
  const int base = blockIdx.x * 256;
  float s0 = 0.f, s1 = 0.f, s2 = 0.f, s3 = 0.f;
  for (int m = 0; m < 256; ++m) {
    const float* Xm = X + (size_t)(base + m) * 1024;
    s0 += Xm[tid];       s1 += Xm[tid + 256];
    s2 += Xm[tid + 512]; s3 += Xm[tid + 768];
  }
  atomicAdd(&Gacc[tid], s0);       atomicAdd(&Gacc[tid + 256], s1);
  atomicAdd(&Gacc[tid + 512], s2); atomicAdd(&Gacc[tid + 768], s3);
}

// Single wave: G^{1/2}, G^{-1/2} of the arithmetic mean (stored paired-row).
__global__ void kPrepG(float* ws) {
  __shared__ float Ab[1024], Yb[1024], Zb[1024], Tb[1024], Ub[1024];
  const int lane = threadIdx.x & 31;
  float *Y = Yb, *Z = Zb, *T = Tb, *U = Ub;
  const float inv_m = 1.0f / (float)NMAT;
  for (int r = 0; r < 32; ++r)
    Ab[mi(r, lane)] = ws[WS_GACC + r * 32 + lane] * inv_m;  // remap in
  float c;
  wave_ns(Ab, Y, Z, T, U, lane, 12, c);
  const float sc = sqrtf(c), isc = 1.0f / sc;
  for (int i = lane; i < 1024; i += 32) {                   // identity copy out
    ws[WS_GSQ + i]  = Y[i] * sc;
    ws[WS_GISQ + i] = Z[i] * isc;
  }
}

// Batched: L += log(G^{-1/2} X G^{-1/2}) via inverse scaling-and-squaring.
// 2 waves/block, one matrix per wave, all work resident in LDS (48 KB).
__global__ void __launch_bounds__(64) kLogAccum(const float* __restrict__ X,
                                                const float* __restrict__ ws,
                                                float* Lacc) {
  __shared__ float Gis[1024];
  __shared__ float lsum[1024];
  __shared__ float buf[2][5][1024];
  const int tid = threadIdx.x;
  const int wave = tid >> 5, lane = tid & 31;
  for (int i = tid; i < 1024; i += 64) { Gis[i] = ws[WS_GISQ + i]; lsum[i] = 0.0f; }
  __syncthreads();

  const int mat = blockIdx.x * 2 + wave;
  const float* Xm = X + (size_t)mat * 1024;
  float* A = buf[wave][0];
  float* Y = buf[wave][1];
  float* Z = buf[wave][2];
  float* T = buf[wave][3];
  float* U = buf[wave][4];

  for (int r = 0; r < 32; ++r) A[mi(r, lane)] = Xm[r * 32 + lane];
  // whiten: A <- Gis * X * Gis
  wave_mm32<false>(Gis, A, Y, lane);
  wave_mm32<false>(Y, Gis, A, lane);

  // s = 3 successive Newton-Schulz square roots: A <- A^{1/8}
  for (int stage = 0; stage < 3; ++stage) {
    float c;
    wave_ns(A, Y, Z, T, U, lane, 7, c);
    const float sc = sqrtf(c);
    for (int i = lane; i < 1024; i += 32) A[i] = Y[i] * sc;
  }

  // log(I + S), S = A - I, Horner; diag-adds as per-lane fixups
  for (int r = 0; r < 32; ++r)
    T[mi(r, lane)] = A[mi(r, lane)] - ((r == lane) ? 1.0f : 0.0f);
  const int MT = 8;
  {
    const float cm = ((MT & 1) ? 1.0f : -1.0f) / (float)MT;
    for (int r = 0; r < 32; ++r) Y[mi(r, lane)] = (r == lane) ? cm : 0.0f;
  }
  for (int k = MT - 1; k >= 1; --k) {
    const float ck = ((k & 1) ? 1.0f : -1.0f) / (float)k;
    wave_mm32e<false, EPI_STORE, true>(T, Y, U, lane, 1.0f, ck); // U=S*Y+ck*I
    { float* t = Y; Y = U; U = t; }
  }
  // lsum += 8 * S*Y  (2^3 undoes the three square roots), atomic epilogue
  wave_mm32e<false, EPI_ATOMIC, false>(T, Y, lsum, lane, 8.0f, 0.0f);
  __syncthreads();
  for (int i = tid; i < 1024; i += 64) atomicAdd(&Lacc[i], lsum[i]);
}

// Single wave: E = exp(L/M); Gf = G^{1/2} E G^{1/2}; P = W^{1/2} Gf^{-1/2}.
__global__ void kPrepP(const float* __restrict__ weight, float* ws) {
  __shared__ float Ab[1024], Yb[1024], Zb[1024], Tb[1024], Ub[1024], G2[1024];
  const int lane = threadIdx.x & 31;
  float *A = Ab, *Y = Yb, *Z = Zb, *T = Tb, *U = Ub;
  const float inv_m = 1.0f / (float)NMAT;

  // S = L/16 ; exp(S) via Horner (8 terms, diag fixup), then 4 squarings
  for (int i = lane; i < 1024; i += 32)
    T[i] = ws[WS_LACC + i] * inv_m * (1.0f / 16.0f);
  for (int r = 0; r < 32; ++r) Y[mi(r, lane)] = (r == lane) ? 1.0f : 0.0f;
  for (int k = 8; k >= 1; --k) {
    wave_mm32e<false, EPI_STORE, true>(T, Y, U, lane, 1.0f / (float)k, 1.0f);
    float* t = Y; Y = U; U = t;
  }
  for (int sq = 0; sq < 4; ++sq) {
    wave_mm32<false>(Y, Y, U, lane);
    float* t = Y; Y = U; U = t;
  }

  // Gf = G^{1/2} * E * G^{1/2}
  for (int i = lane; i < 1024; i += 32) G2[i] = ws[WS_GSQ + i];
  wave_mm32<false>(G2, Y, Z, lane);
  wave_mm32<false>(Z, G2, A, lane);

  // Gf^{-1/2} -> G2
  float c;
  wave_ns(A, Y, Z, T, U, lane, 12, c);
  const float isc = 1.0f / sqrtf(c);
  for (int i = lane; i < 1024; i += 32) G2[i] = Z[i] * isc;

  // W^{1/2}: NS on weight, then P = sqrt(cw) * (Y * Gf^{-1/2}) via alpha
  for (int r = 0; r < 32; ++r) A[mi(r, lane)] = weight[r * 32 + lane];
  float cw;
  wave_ns(A, Y, Z, T, U, lane, 12, cw);
  wave_mm32e<false, EPI_STORE, false>(Y, G2, U, lane, sqrtf(cw), 0.0f);
  for (int i = lane; i < 1024; i += 32) ws[WS_P + i] = U[i];
}

// Batched: out = P X P^T; result written straight to global (row-major)
// from the WMMA epilogue.
__global__ void __launch_bounds__(128) kFinal(const float* __restrict__ X,
                                              const float* __restrict__ ws,
                                              float* __restrict__ out) {
  __shared__ float P[1024];
  __shared__ float buf[4][2][1024];
  const int tid = threadIdx.x;
  const int wave = tid >> 5, lane = tid & 31;
  for (int i = tid; i < 1024; i += 128) P[i] = ws[WS_P + i];
  __syncthreads();
  const int mat = blockIdx.x * 4 + wave;
  const float* Xm = X + (size_t)mat * 1024;
  float* B0 = buf[wave][0];
  float* B1 = buf[wave][1];
  for (int r = 0; r < 32; ++r) B0[mi(r, lane)] = Xm[r * 32 + lane];
  wave_mm32<false>(P, B0, B1, lane);                       // T1 = P X
  wave_mm32e<true, EPI_GSTORE, false>(B1, P, out + (size_t)mat * 1024,
                                      lane, 1.0f, 0.0f);   // out = T1 P^T
}

// ---------------------------------------------------------------------------
extern "C" void kernel_launch(void* const* d_in, const int* in_sizes, int n_in,
                              void* d_out, int out_size, void* d_ws, size_t ws_size,
                              hipStream_t stream) {
  const float* X = (const float*)d_in[0];       // (256,64,32,32) fp32
  const float* W = (const float*)d_in[1];       // (32,32) fp32
  float* out = (float*)d_out;
  float* ws = (float*)d_ws;
  (void)in_sizes; (void)n_in; (void)out_size; (void)ws_size;

  kZero<<<8, 256, 0, stream>>>(ws);
  kMean<<<64, 256, 0, stream>>>(X, ws + WS_GACC);
  kPrepG<<<1, 32, 0, stream>>>(ws);
  kLogAccum<<<NMAT / 2, 64, 0, stream>>>(X, ws, ws + WS_LACC);
  kPrepP<<<1, 32, 0, stream>>>(W, ws);
  kFinal<<<NMAT / 4, 128, 0, stream>>>(X, ws, out);
}